// MixModel_70892730188006
// MI455X (gfx1250) — compile-verified
//
#include <hip/hip_runtime.h>
#include <hip/hip_bf16.h>
#include <cstdint>
#include <cstddef>

// ---------------- model dims ----------------
#define N_NODES 50000
#define N_EDGES 800000
#define E_TOT   (N_EDGES + N_NODES)   // with self loops
#define DIM_IN  128
#define HIDC    64
#define HEADS   4
#define HCC     256                   // HEADS*HID
#define OUT_DIM 40

typedef __attribute__((ext_vector_type(2))) float v2f;
typedef __attribute__((ext_vector_type(8))) float v8f;

static __host__ __device__ inline int cdiv_i(int a, int b) { return (a + b - 1) / b; }

// ---------------- generic WMMA f32 GEMM ----------------
// C[50000,M] = op(A[50000,K] @ B[K,M] (+bias) (+C)), optional relu.
// One wave computes a 80x16 strip: 5 row tiles (3125 = 5^5 row tiles total,
// so 5 divides exactly -> no row guard, EXEC all-1s at every WMMA).
// Each k-step loads one B fragment and feeds 5 V_WMMA_F32_16X16X4_F32.
template <int K, int M, bool RELU, bool BIAS, bool ACCUM, bool COLMASK>
__global__ void gemm_wmma_f32(const float* __restrict__ A,
                              const float* __restrict__ B,
                              const float* __restrict__ bias,
                              float* __restrict__ C)
{
    constexpr int colTiles  = (M + 15) / 16;
    constexpr int ROWB      = 5;                      // row tiles per wave
    constexpr int rowBlocks = (N_NODES / 16) / ROWB;  // 625

    const int tile = blockIdx.x * blockDim.y + threadIdx.y;   // wave-uniform
    if (tile >= rowBlocks * colTiles) return;

    const int rb   = tile / colTiles;
    const int ct   = tile % colTiles;
    const int lane = threadIdx.x;        // 0..31
    const int half = lane >> 4;          // 0 or 1
    const int l16  = lane & 15;

    const int row0 = rb * (ROWB * 16);
    const int colB = ct * 16 + l16;
    const bool colOK = !COLMASK || (colB < M);

    const float* Ar[ROWB];
    #pragma unroll
    for (int t = 0; t < ROWB; ++t)
        Ar[t] = A + (size_t)(row0 + t * 16 + l16) * K;

    const float* Bp = B + (colOK ? colB : 0);   // lane's B column (safe base)
    v8f acc[ROWB] = {};

    for (int k0 = 0; k0 < K; k0 += 4) {
        // lanes 0-15 -> K=k0,k0+1 ; lanes 16-31 -> K=k0+2,k0+3
        const int ka = k0 + 2 * half;
        v2f b;
        if (COLMASK && !colOK) {
            b.x = 0.0f; b.y = 0.0f;
        } else {
            b.x = Bp[(size_t)ka * M];
            b.y = Bp[(size_t)(ka + 1) * M];
        }
        #pragma unroll
        for (int t = 0; t < ROWB; ++t) {
            v2f a;
            a.x = Ar[t][ka];            // contiguous pair -> global_load_b64
            a.y = Ar[t][ka + 1];
            acc[t] = __builtin_amdgcn_wmma_f32_16x16x4_f32(
                false, a, false, b, (short)0, acc[t], false, false);
        }
    }

    float bv = 0.0f;
    if (BIAS && colOK) bv = bias[colB];

    #pragma unroll
    for (int t = 0; t < ROWB; ++t) {
        #pragma unroll
        for (int i = 0; i < 8; ++i) {
            // C/D layout: VGPR i -> row i (lanes 0-15) / row 8+i (lanes 16-31)
            const int r = row0 + t * 16 + i + 8 * half;
            if (colOK) {
                float v = acc[t][i];
                if (BIAS)  v += bv;
                if (ACCUM) v += C[(size_t)r * M + colB];
                if (RELU)  v = fmaxf(v, 0.0f);
                C[(size_t)r * M + colB] = v;
            }
        }
    }
}

// ---------------- helpers ----------------
__global__ void fill_kernel(float* __restrict__ p, float v, int n)
{
    int i = blockIdx.x * blockDim.x + threadIdx.x;
    if (i < n) p[i] = v;
}

__device__ inline void atomicMaxF(float* addr, float val)
{
    int* ia = (int*)addr;
    int old = *ia;
    while (__int_as_float(old) < val) {
        int assumed = old;
        old = atomicCAS(ia, assumed, __float_as_int(val));
        if (old == assumed) break;
    }
}

// ---------------- GATv2 edge kernels ----------------
// E1: per-edge scores e[h] = sum_c att[h,c] * leaky_relu(xl[src,h,c] + xr[dst,h,c])
// and segment max over dst. One wave per edge (32 lanes across 64 channels/head).
__global__ void gat_edge_scores(const int* __restrict__ ei,
                                const float* __restrict__ xl,
                                const float* __restrict__ xr,
                                const float* __restrict__ att,
                                float* __restrict__ esc,
                                float* __restrict__ emax)
{
    const int e = blockIdx.x * blockDim.y + threadIdx.y;   // wave-uniform
    if (e >= E_TOT) return;
    const int lane = threadIdx.x;
    int src, dst;
    if (e < N_EDGES) { src = ei[e]; dst = ei[N_EDGES + e]; }
    else             { src = e - N_EDGES; dst = src; }

    const float* pl = xl + (size_t)src * HCC;
    const float* pr = xr + (size_t)dst * HCC;

    for (int h = 0; h < HEADS; ++h) {
        const int c0 = h * HIDC + lane;
        float m0 = pl[c0]      + pr[c0];
        float m1 = pl[c0 + 32] + pr[c0 + 32];
        m0 = (m0 > 0.0f) ? m0 : 0.2f * m0;
        m1 = (m1 > 0.0f) ? m1 : 0.2f * m1;
        float p = m0 * att[h * HIDC + lane] + m1 * att[h * HIDC + lane + 32];
        for (int off = 16; off > 0; off >>= 1) p += __shfl_xor(p, off, 32);
        if (lane == 0) {
            esc[(size_t)e * HEADS + h] = p;
            atomicMaxF(&emax[dst * HEADS + h], p);
        }
    }
}

// E2: ee = exp(e - emax[dst]); denom[dst] += ee   (thread per edge)
__global__ void gat_edge_expsum(const int* __restrict__ ei,
                                float* __restrict__ esc,
                                const float* __restrict__ emax,
                                float* __restrict__ denom)
{
    const int e = blockIdx.x * blockDim.x + threadIdx.x;
    if (e >= E_TOT) return;
    const int dst = (e < N_EDGES) ? ei[N_EDGES + e] : (e - N_EDGES);
    for (int h = 0; h < HEADS; ++h) {
        float ee = __expf(esc[(size_t)e * HEADS + h] - emax[dst * HEADS + h]);
        esc[(size_t)e * HEADS + h] = ee;
        atomicAdd(&denom[dst * HEADS + h], ee);
    }
}

// E3: g[dst] += alpha * xl[src]   (one wave per edge, 256 channels)
__global__ void gat_edge_aggregate(const int* __restrict__ ei,
                                   const float* __restrict__ xl,
                                   const float* __restrict__ esc,
                                   const float* __restrict__ denom,
                                   float* __restrict__ gacc)
{
    const int e = blockIdx.x * blockDim.y + threadIdx.y;   // wave-uniform
    if (e >= E_TOT) return;
    const int lane = threadIdx.x;
    int src, dst;
    if (e < N_EDGES) { src = ei[e]; dst = ei[N_EDGES + e]; }
    else             { src = e - N_EDGES; dst = src; }

    const float* pl = xl + (size_t)src * HCC;
    float* pd = gacc + (size_t)dst * HCC;
    for (int h = 0; h < HEADS; ++h) {
        const float alpha = esc[(size_t)e * HEADS + h] /
                            (denom[dst * HEADS + h] + 1e-16f);
        const int c = h * HIDC + lane;
        atomicAdd(&pd[c],      alpha * pl[c]);
        atomicAdd(&pd[c + 32], alpha * pl[c + 32]);
    }
}

// ---------------- GroupNorm kernels ----------------
// gn1: g = groupnorm(relu(gacc + gat_b)), C=256, 16 groups of 16, in-place.
__global__ void gn1_kernel(float* __restrict__ g,
                           const float* __restrict__ gat_b,
                           const float* __restrict__ w,
                           const float* __restrict__ b)
{
    const int idx = blockIdx.x * blockDim.x + threadIdx.x;
    if (idx >= N_NODES * 16) return;
    const int r = idx >> 4, grp = idx & 15;
    float* p = g + (size_t)r * HCC + grp * 16;
    float v[16];
    float mu = 0.0f;
    #pragma unroll
    for (int i = 0; i < 16; ++i) {
        float t = fmaxf(p[i] + gat_b[grp * 16 + i], 0.0f);
        v[i] = t; mu += t;
    }
    mu *= (1.0f / 16.0f);
    float var = 0.0f;
    #pragma unroll
    for (int i = 0; i < 16; ++i) { float d = v[i] - mu; var += d * d; }
    var *= (1.0f / 16.0f);
    const float rs = rsqrtf(var + 1e-5f);
    #pragma unroll
    for (int i = 0; i < 16; ++i)
        p[i] = (v[i] - mu) * rs * w[grp * 16 + i] + b[grp * 16 + i];
}

// gn2: out = groupnorm(relu(s)) + res, C=64, 16 groups of 4.
__global__ void gn2_kernel(const float* __restrict__ s,
                           const float* __restrict__ w,
                           const float* __restrict__ b,
                           const float* __restrict__ res,
                           float* __restrict__ out)
{
    const int idx = blockIdx.x * blockDim.x + threadIdx.x;
    if (idx >= N_NODES * 16) return;
    const int r = idx >> 4, grp = idx & 15;
    const float* p = s + (size_t)r * HIDC + grp * 4;
    float v[4];
    float mu = 0.0f;
    #pragma unroll
    for (int i = 0; i < 4; ++i) { float t = fmaxf(p[i], 0.0f); v[i] = t; mu += t; }
    mu *= 0.25f;
    float var = 0.0f;
    #pragma unroll
    for (int i = 0; i < 4; ++i) { float d = v[i] - mu; var += d * d; }
    var *= 0.25f;
    const float rs = rsqrtf(var + 1e-5f);
    #pragma unroll
    for (int i = 0; i < 4; ++i) {
        const int c = grp * 4 + i;
        out[(size_t)r * HIDC + c] =
            (v[i] - mu) * rs * w[c] + b[c] + res[(size_t)r * HIDC + c];
    }
}

// ---------------- SAGE kernels ----------------
// one wave per original edge, 128 channels
__global__ void sage_agg(const int* __restrict__ ei,
                         const float* __restrict__ g2,
                         float* __restrict__ msum,
                         float* __restrict__ cnt)
{
    const int e = blockIdx.x * blockDim.y + threadIdx.y;   // wave-uniform
    if (e >= N_EDGES) return;
    const int lane = threadIdx.x;
    const int src = ei[e], dst = ei[N_EDGES + e];
    const float* ps = g2 + (size_t)src * 128;
    float* pd = msum + (size_t)dst * 128;
    atomicAdd(&pd[lane],      ps[lane]);
    atomicAdd(&pd[lane + 32], ps[lane + 32]);
    atomicAdd(&pd[lane + 64], ps[lane + 64]);
    atomicAdd(&pd[lane + 96], ps[lane + 96]);
    if (lane == 0) atomicAdd(&cnt[dst], 1.0f);
}

__global__ void sage_mean(float* __restrict__ msum, const float* __restrict__ cnt)
{
    const int i = blockIdx.x * blockDim.x + threadIdx.x;
    if (i >= N_NODES * 128) return;
    msum[i] /= fmaxf(cnt[i >> 7], 1.0f);
}

// ---------------- log-softmax over 40 logits ----------------
__global__ void log_softmax_kernel(const float* __restrict__ logits,
                                   float* __restrict__ out)
{
    const int r = blockIdx.x * blockDim.x + threadIdx.x;
    if (r >= N_NODES) return;
    const float* p = logits + (size_t)r * OUT_DIM;
    float mx = p[0];
    for (int j = 1; j < OUT_DIM; ++j) mx = fmaxf(mx, p[j]);
    float s = 0.0f;
    for (int j = 0; j < OUT_DIM; ++j) s += __expf(p[j] - mx);
    const float ls = __logf(s);
    for (int j = 0; j < OUT_DIM; ++j)
        out[(size_t)r * OUT_DIM + j] = p[j] - mx - ls;
}

// ---------------- host orchestration ----------------
extern "C" void kernel_launch(void* const* d_in, const int* in_sizes, int n_in,
                              void* d_out, int out_size, void* d_ws, size_t ws_size,
                              hipStream_t stream)
{
    (void)in_sizes; (void)n_in; (void)out_size; (void)ws_size;

    const float* x       = (const float*)d_in[0];
    const int*   ei      = (const int*)  d_in[1];
    const float* w1      = (const float*)d_in[2];
    const float* b1      = (const float*)d_in[3];
    const float* gat_wl  = (const float*)d_in[4];
    const float* gat_wr  = (const float*)d_in[5];
    const float* gat_att = (const float*)d_in[6];
    const float* gat_b   = (const float*)d_in[7];
    const float* g1_w    = (const float*)d_in[8];
    const float* g1_b    = (const float*)d_in[9];
    const float* w3      = (const float*)d_in[10];
    const float* b3      = (const float*)d_in[11];
    const float* sage_wl = (const float*)d_in[12];
    const float* sage_bl = (const float*)d_in[13];
    const float* sage_wr = (const float*)d_in[14];
    const float* g2_w    = (const float*)d_in[15];
    const float* g2_b    = (const float*)d_in[16];
    const float* w2      = (const float*)d_in[17];
    const float* b2      = (const float*)d_in[18];
    const float* w4      = (const float*)d_in[19];
    const float* b4      = (const float*)d_in[20];
    float* out = (float*)d_out;

    // workspace carve-out (float units, 1KB-aligned chunks)
    float* ws = (float*)d_ws;
    size_t off = 0;
    auto alloc = [&](size_t n) {
        float* p = ws + off;
        off += (n + 255) & ~(size_t)255;
        return p;
    };
    float* hbuf  = alloc((size_t)N_NODES * HIDC);    // h (also residual)
    float* xl    = alloc((size_t)N_NODES * HCC);     // reused for s2 + logits later
    float* xr    = alloc((size_t)N_NODES * HCC);     // reused as gacc after E1
    float* esc   = alloc((size_t)E_TOT * HEADS);
    float* emax  = alloc((size_t)N_NODES * HEADS);
    float* denom = alloc((size_t)N_NODES * HEADS);
    float* g2    = alloc((size_t)N_NODES * 128);
    float* msum  = alloc((size_t)N_NODES * 128);
    float* cnt   = alloc((size_t)N_NODES);
    float* sbuf  = alloc((size_t)N_NODES * HIDC);
    float* gn2   = alloc((size_t)N_NODES * HIDC);
    float* gacc  = xr;                               // alias: xr dead after E1
    float* s2    = xl;                               // alias: xl dead after E3
    float* logit = xl + ((size_t)N_NODES * 32 + 256);

    const dim3 gemmBlk(32, 4);
    auto gemmGrid = [](int colTiles) {
        return dim3((unsigned)cdiv_i(625 * colTiles, 4));
    };
    const dim3 edgeBlk(32, 8);

    // init accumulators
    fill_kernel<<<cdiv_i(N_NODES * HEADS, 256), 256, 0, stream>>>(
        emax, -3.402823466e38f, N_NODES * HEADS);
    hipMemsetAsync(denom, 0, (size_t)N_NODES * HEADS * sizeof(float), stream);
    hipMemsetAsync(msum,  0, (size_t)N_NODES * 128 * sizeof(float), stream);
    hipMemsetAsync(cnt,   0, (size_t)N_NODES * sizeof(float), stream);

    // 1) h = relu(x @ w1 + b1)
    gemm_wmma_f32<128, 64, true, true, false, false>
        <<<gemmGrid(4), gemmBlk, 0, stream>>>(x, w1, b1, hbuf);

    // 2) xl = h @ gat_wl ; xr = h @ gat_wr
    gemm_wmma_f32<64, 256, false, false, false, false>
        <<<gemmGrid(16), gemmBlk, 0, stream>>>(hbuf, gat_wl, nullptr, xl);
    gemm_wmma_f32<64, 256, false, false, false, false>
        <<<gemmGrid(16), gemmBlk, 0, stream>>>(hbuf, gat_wr, nullptr, xr);

    // 3) GATv2 edge phase
    gat_edge_scores<<<cdiv_i(E_TOT, 8), edgeBlk, 0, stream>>>(ei, xl, xr, gat_att,
                                                              esc, emax);
    hipMemsetAsync(gacc, 0, (size_t)N_NODES * HCC * sizeof(float), stream);
    gat_edge_expsum<<<cdiv_i(E_TOT, 256), 256, 0, stream>>>(ei, esc, emax, denom);
    gat_edge_aggregate<<<cdiv_i(E_TOT, 8), edgeBlk, 0, stream>>>(ei, xl, esc,
                                                                 denom, gacc);

    // 4) g = groupnorm(relu(g + gat_b)) (in place), then g2 = relu(g @ w3 + b3)
    gn1_kernel<<<cdiv_i(N_NODES * 16, 256), 256, 0, stream>>>(gacc, gat_b, g1_w, g1_b);
    gemm_wmma_f32<256, 128, true, true, false, false>
        <<<gemmGrid(8), gemmBlk, 0, stream>>>(gacc, w3, b3, g2);

    // 5) SAGE: mean aggregation + two GEMMs into s
    sage_agg<<<cdiv_i(N_EDGES, 8), edgeBlk, 0, stream>>>(ei, g2, msum, cnt);
    sage_mean<<<cdiv_i(N_NODES * 128, 256), 256, 0, stream>>>(msum, cnt);
    gemm_wmma_f32<128, 64, false, true, false, false>
        <<<gemmGrid(4), gemmBlk, 0, stream>>>(msum, sage_wl, sage_bl, sbuf);
    gemm_wmma_f32<128, 64, false, false, true, false>
        <<<gemmGrid(4), gemmBlk, 0, stream>>>(g2, sage_wr, nullptr, sbuf);

    // 6) s = groupnorm(relu(s)) + res
    gn2_kernel<<<cdiv_i(N_NODES * 16, 256), 256, 0, stream>>>(sbuf, g2_w, g2_b,
                                                              hbuf, gn2);

    // 7) s2 = relu(gn2 @ w2 + b2) ; logits = s2 @ w4 + b4
    gemm_wmma_f32<64, 32, true, true, false, false>
        <<<gemmGrid(2), gemmBlk, 0, stream>>>(gn2, w2, b2, s2);
    gemm_wmma_f32<32, 40, false, true, false, true>
        <<<gemmGrid(3), gemmBlk, 0, stream>>>(s2, w4, b4, logit);

    // 8) log_softmax
    log_softmax_kernel<<<cdiv_i(N_NODES, 256), 256, 0, stream>>>(logit, out);
}